// CG_11682311045589
// MI455X (gfx1250) — compile-verified
//
#include <hip/hip_runtime.h>
#include <math.h>

// Parzen (cubic B-spline) histogram density lookup, B=2, N=2, V=96^3, K=20.
// Per-voxel independent -> HBM-roofline streaming kernel (~42 MB single pass
// => ~1.8 us at 23.3 TB/s). No contraction dim => WMMA inapplicable; optimal
// CDNA5 path is b128 VMEM + branchless VALU.

#define NUM_BINS 16
#define RADIUS   2
#define KBINS    (NUM_BINS + 2 * RADIUS)   // 20
#define EPSF     1e-8f

// Cubic B-spline via clamped-cube identity (exact for all d):
//   w(d) = ( max(2-|d|,0)^3 - 4*max(1-|d|,0)^3 ) / 6
// Branchless: no v_cmp/v_cndmask, no VCC serialization.
__device__ __forceinline__ float bspline(float d) {
    float ad = fabsf(d);
    float t2 = fmaxf(2.0f - ad, 0.0f);
    float t1 = fmaxf(1.0f - ad, 0.0f);
    float c2 = (t2 * t2) * t2;
    float c1 = (t1 * t1) * t1;
    return fmaf(-4.0f, c1, c2) * (1.0f / 6.0f);
}

// One voxel: two image samples -> two density lookups.
__device__ __forceinline__ void voxel(float x0, float x1, float m0, float m1,
                                      float& o0, float& o1) {
    float mx = fmaxf(x0, x1);
    float mn = fminf(x0, x1);
    float bw = (mx - mn) * (1.0f / (float)NUM_BINS);
    float pad_min = mn - bw * (float)RADIUS;
    float bws = fmaxf(bw, EPSF);               // jnp.clip(bin_width, EPS)

    // IEEE division: floor(bin_pos) lands exactly on integers for the
    // min/max samples; must round identically to the reference.
    float bp0 = (x0 - pad_min) / bws;
    float bp1 = (x1 - pad_min) / bws;
    float f0 = floorf(bp0);
    float f1 = floorf(bp1);

    // clipped bin index -> window base (min_win_idx = bin_idx - (R-1))
    float base0 = fminf(fmaxf(f0, (float)RADIUS), (float)(RADIUS + NUM_BINS - 1)) - 1.0f;
    float base1 = fminf(fmaxf(f1, (float)RADIUS), (float)(RADIUS + NUM_BINS - 1)) - 1.0f;

    // 4-tap window weights per sample; windows always land inside [0,19],
    // so the full tap sum equals the histogram sum over all K bins.
    float w0[2 * RADIUS], w1[2 * RADIUS];
    float total = 0.0f;
#pragma unroll
    for (int r = 0; r < 2 * RADIUS; ++r) {
        w0[r] = bspline(bp0 - (base0 + (float)r));
        w1[r] = bspline(bp1 - (base1 + (float)r));
        total += w0[r] + w1[r];
    }
    float inv_tot = 1.0f / fmaxf(total, EPSF);

    // gather indices: UNclipped floor, clipped to [0, K-1]
    float t0 = fminf(fmaxf(f0, 0.0f), (float)(KBINS - 1));
    float t1 = fminf(fmaxf(f1, 0.0f), (float)(KBINS - 1));

    // hist[t] = sum of taps (from both samples) landing on bin t
    float h0 = 0.0f, h1 = 0.0f;
#pragma unroll
    for (int r = 0; r < 2 * RADIUS; ++r) {
        float k0 = base0 + (float)r;
        float k1 = base1 + (float)r;
        h0 += ((k0 == t0) ? w0[r] : 0.0f) + ((k1 == t0) ? w1[r] : 0.0f);
        h1 += ((k0 == t1) ? w0[r] : 0.0f) + ((k1 == t1) ? w1[r] : 0.0f);
    }

    o0 = h0 * inv_tot * ((m0 != 0.0f) ? 1.0f : 0.0f);
    o1 = h1 * inv_tot * ((m1 != 0.0f) ? 1.0f : 0.0f);
}

// Grid: x covers V/4 voxel-quads, y covers the B=2 batches (no int division).
__global__ __launch_bounds__(256) void parzen_density_kernel(
    const float* __restrict__ img, const float* __restrict__ mask,
    float* __restrict__ out, int Vq, int V) {
    int q = blockIdx.x * blockDim.x + threadIdx.x;
    if (q >= Vq) return;
    int b = blockIdx.y;

    size_t off0 = (size_t)(b * 2 + 0) * (size_t)V + (size_t)q * 4;  // n=0
    size_t off1 = off0 + (size_t)V;                                 // n=1

    // Pull the next stride into L2 ahead of the wave (global_prefetch_b8).
    const size_t PF = 256 * 4 * 8;             // 8 blocks ahead (32 KB)
    size_t limit = (size_t)4 * (size_t)V;      // B*N*V elements
    if (off1 + PF + 4 <= limit) {
        __builtin_prefetch(img + off0 + PF, 0, 1);
        __builtin_prefetch(img + off1 + PF, 0, 1);
    }

    float4 x0 = *(const float4*)(img + off0);
    float4 x1 = *(const float4*)(img + off1);
    float4 mk0 = *(const float4*)(mask + off0);
    float4 mk1 = *(const float4*)(mask + off1);

    float4 y0, y1;
    voxel(x0.x, x1.x, mk0.x, mk1.x, y0.x, y1.x);
    voxel(x0.y, x1.y, mk0.y, mk1.y, y0.y, y1.y);
    voxel(x0.z, x1.z, mk0.z, mk1.z, y0.z, y1.z);
    voxel(x0.w, x1.w, mk0.w, mk1.w, y0.w, y1.w);

    *(float4*)(out + off0) = y0;
    *(float4*)(out + off1) = y1;
}

extern "C" void kernel_launch(void* const* d_in, const int* in_sizes, int n_in,
                              void* d_out, int out_size, void* d_ws, size_t ws_size,
                              hipStream_t stream) {
    (void)n_in; (void)out_size; (void)d_ws; (void)ws_size;
    const float* images = (const float*)d_in[0];   // [2,2,1,96,96,96] f32
    const float* mask   = (const float*)d_in[1];   // [2,2,96,96,96]   f32
    float* out = (float*)d_out;                    // [2,2,96,96,96]   f32

    int V  = in_sizes[0] / 4;                      // B*N = 4 images
    int Vq = V / 4;                                // 96^3 divisible by 4
    int threads = 256;
    dim3 grid((Vq + threads - 1) / threads, 2);    // y = batch
    parzen_density_kernel<<<grid, threads, 0, stream>>>(images, mask, out, Vq, V);
}